// GCN_19404662243570
// MI455X (gfx1250) — compile-verified
//
#include <hip/hip_runtime.h>
#include <hip/hip_bf16.h>

typedef float v2f __attribute__((ext_vector_type(2)));
typedef float v8f __attribute__((ext_vector_type(8)));

#define FEAT1 256   // N_FEAT
#define HID   128   // HIDDEN
#define NCLS  16    // N_CLASSES

// ---------------------------------------------------------------------------
// Degree kernels: deg starts at 1.0 (self loop), count dst, then rsqrt in place
// ---------------------------------------------------------------------------
__global__ __launch_bounds__(256) void k_deg_init(float* __restrict__ deg, int n) {
    int i = blockIdx.x * 256 + threadIdx.x;
    if (i < n) deg[i] = 1.0f;
}

__global__ __launch_bounds__(256) void k_deg_count(const int* __restrict__ dst,
                                                   float* __restrict__ deg, int e) {
    int i = blockIdx.x * 256 + threadIdx.x;
    if (i < e) atomicAdd(&deg[dst[i]], 1.0f);
}

__global__ __launch_bounds__(256) void k_deg_fin(float* __restrict__ deg, int n) {
    int i = blockIdx.x * 256 + threadIdx.x;
    if (i < n) deg[i] = __frsqrt_rn(deg[i]);   // deg >= 1 always (self loop)
}

// ---------------------------------------------------------------------------
// WMMA GEMM: C[M x N] = A[M x K] * B[K x N], row-major, f32.
// Block = 256 threads (8 waves). Block owns 16 rows; wave w owns cols [16w,16w+16).
// A tile DMA'd straight into LDS via GLOBAL_LOAD_ASYNC_TO_LDS_B128 (ASYNCcnt),
// padded by +4 floats/row to dodge 16-way LDS bank conflicts.
// Requires: M % 16 == 0, K % 16 == 0, N == 128, K <= 256. No divergence near WMMA.
// ---------------------------------------------------------------------------
__global__ __launch_bounds__(256) void k_gemm16(const float* __restrict__ A,
                                                const float* __restrict__ B,
                                                float* __restrict__ C,
                                                int K, int N) {
    __shared__ float As[16 * (256 + 4)];
    const int tid  = threadIdx.x;
    const int row0 = blockIdx.x * 16;
    const int KS   = K + 4;
    const int Kq4  = K >> 2;                    // float4 chunks per row

    // async memory->LDS DMA of the 16 x K A-tile, 16 bytes per lane per issue.
    // 16*Kq4 is a multiple of 256, so the loop has no divergent tail.
    for (int idx = tid; idx < 16 * Kq4; idx += 256) {
        int r  = idx / Kq4;
        int c4 = (idx - r * Kq4) << 2;
        unsigned lds = (unsigned)(size_t)&As[r * KS + c4];           // low 32b = LDS offset
        unsigned long long ga =
            (unsigned long long)(const void*)&A[(size_t)(row0 + r) * K + c4];
        asm volatile("global_load_async_to_lds_b128 %0, %1, off"
                     :: "v"(lds), "v"(ga) : "memory");
    }
    asm volatile("s_wait_asynccnt 0x0" ::: "memory");
    __syncthreads();

    const int wave = tid >> 5;
    const int lane = tid & 31;
    const int half = lane >> 4;     // 0: K-offsets {0,1}; 1: K-offsets {2,3}
    const int l16  = lane & 15;     // A: row M; B/C: col N
    const int n0   = wave * 16;

    const float* ap = &As[l16 * KS + half * 2];
    const float* bp = &B[(size_t)(half * 2) * N + n0 + l16];

    v8f acc = {};
    for (int k0 = 0; k0 < K; k0 += 4) {
        v2f a, b;
        a.x = ap[k0];
        a.y = ap[k0 + 1];
        b.x = bp[0];
        b.y = bp[N];
        acc = __builtin_amdgcn_wmma_f32_16x16x4_f32(
            /*neg_a=*/false, a, /*neg_b=*/false, b,
            /*c_mod=*/(short)0, acc, /*reuse_a=*/false, /*reuse_b=*/false);
        bp += 4 * N;
    }

    // C/D layout: VGPR j -> M = j + 8*half, N = l16
    #pragma unroll
    for (int j = 0; j < 8; ++j) {
        int m = j + half * 8;
        C[(size_t)(row0 + m) * N + n0 + l16] = acc[j];
    }
}

// ---------------------------------------------------------------------------
// Aggregation: out = dis[i]^2 * F[i]  (self-loop term), then edge scatter-add,
// then fused bias + ReLU. Feature dim fixed at HID=128.
// ---------------------------------------------------------------------------
__global__ __launch_bounds__(256) void k_agg_init(const float* __restrict__ F,
                                                  const float* __restrict__ dis,
                                                  float* __restrict__ out, long long total) {
    long long idx = (long long)blockIdx.x * 256 + threadIdx.x;
    if (idx < total) {
        int node = (int)(idx >> 7);
        float di = dis[node];
        out[idx] = F[idx] * di * di;
    }
}

// one wave per edge: 32 lanes x float4 = 128 features; coalesced gather,
// global_atomic_add_f32 scatter
__global__ __launch_bounds__(256) void k_agg_edges(const float* __restrict__ F,
                                                   const int* __restrict__ src,
                                                   const int* __restrict__ dst,
                                                   const float* __restrict__ dis,
                                                   float* __restrict__ out, int e) {
    int wave = (int)((blockIdx.x * 256 + threadIdx.x) >> 5);
    int lane = threadIdx.x & 31;
    if (wave >= e) return;                     // wave-uniform exit
    int s = src[wave];
    int d = dst[wave];
    float norm = dis[s] * dis[d];
    const float4 v = *(const float4*)&F[(size_t)s * HID + lane * 4];
    float* o = &out[(size_t)d * HID + lane * 4];
    atomicAdd(o + 0, v.x * norm);
    atomicAdd(o + 1, v.y * norm);
    atomicAdd(o + 2, v.z * norm);
    atomicAdd(o + 3, v.w * norm);
}

__global__ __launch_bounds__(256) void k_bias_relu(float* __restrict__ h,
                                                   const float* __restrict__ b,
                                                   long long total) {
    long long idx = (long long)blockIdx.x * 256 + threadIdx.x;
    if (idx < total) {
        int c = (int)(idx & (HID - 1));
        float v = h[idx] + b[c];
        h[idx] = v > 0.0f ? v : 0.0f;
    }
}

// ---------------------------------------------------------------------------
// Output head: logits = h @ W_out + b_out, row softmax. One thread per node
// (128x16 = 2048 FMA/thread; W_out/b_out broadcast from cache).
// ---------------------------------------------------------------------------
__global__ __launch_bounds__(256) void k_out_softmax(const float* __restrict__ H,
                                                     const float* __restrict__ Wout,
                                                     const float* __restrict__ bout,
                                                     float* __restrict__ out, int n) {
    int node = blockIdx.x * 256 + threadIdx.x;
    if (node >= n) return;
    float logit[NCLS];
    #pragma unroll
    for (int c = 0; c < NCLS; ++c) logit[c] = bout[c];
    const float* h = &H[(size_t)node * HID];
    for (int k = 0; k < HID; ++k) {
        float hv = h[k];
        #pragma unroll
        for (int c = 0; c < NCLS; ++c) logit[c] = fmaf(hv, Wout[k * NCLS + c], logit[c]);
    }
    float m = logit[0];
    #pragma unroll
    for (int c = 1; c < NCLS; ++c) m = fmaxf(m, logit[c]);
    float sum = 0.0f;
    #pragma unroll
    for (int c = 0; c < NCLS; ++c) { logit[c] = __expf(logit[c] - m); sum += logit[c]; }
    float inv = 1.0f / sum;
    #pragma unroll
    for (int c = 0; c < NCLS; ++c) out[(size_t)node * NCLS + c] = logit[c] * inv;
}

// ---------------------------------------------------------------------------
extern "C" void kernel_launch(void* const* d_in, const int* in_sizes, int n_in,
                              void* d_out, int out_size, void* d_ws, size_t ws_size,
                              hipStream_t stream) {
    const float* x    = (const float*)d_in[0];
    const int*   ei   = (const int*)d_in[1];   // [2, E] row-major int32
    const float* W1   = (const float*)d_in[2];
    const float* b1   = (const float*)d_in[3];
    const float* W2   = (const float*)d_in[4];
    const float* b2   = (const float*)d_in[5];
    const float* Wout = (const float*)d_in[6];
    const float* bout = (const float*)d_in[7];
    float* out = (float*)d_out;

    const int N = in_sizes[0] / FEAT1;         // 50000
    const int E = in_sizes[1] / 2;             // 800000
    const int* src = ei;
    const int* dst = ei + E;

    // workspace layout: dis[N] | bufA[N*128] | bufB[N*128]
    size_t offA = ((size_t)N * sizeof(float) + 255) & ~(size_t)255;
    float* dis  = (float*)d_ws;
    float* bufA = (float*)((char*)d_ws + offA);
    float* bufB = bufA + (size_t)N * HID;

    const long long totNH = (long long)N * HID;
    const int gNH = (int)((totNH + 255) / 256);
    const int gN  = (N + 255) / 256;
    const int gE  = (E + 255) / 256;
    const int gEw = (E + 7) / 8;               // 8 edge-waves per 256-thread block

    // normalization coefficients
    k_deg_init<<<gN, 256, 0, stream>>>(dis, N);
    k_deg_count<<<gE, 256, 0, stream>>>(dst, dis, E);
    k_deg_fin<<<gN, 256, 0, stream>>>(dis, N);

    // layer 1: h = relu(Agg(x @ W1) + b1)
    k_gemm16<<<N / 16, 256, 0, stream>>>(x, W1, bufA, FEAT1, HID);
    k_agg_init<<<gNH, 256, 0, stream>>>(bufA, dis, bufB, totNH);
    k_agg_edges<<<gEw, 256, 0, stream>>>(bufA, src, dst, dis, bufB, E);
    k_bias_relu<<<gNH, 256, 0, stream>>>(bufB, b1, totNH);

    // layer 2: h = relu(Agg(h @ W2) + b2)
    k_gemm16<<<N / 16, 256, 0, stream>>>(bufB, W2, bufA, HID, HID);
    k_agg_init<<<gNH, 256, 0, stream>>>(bufA, dis, bufB, totNH);
    k_agg_edges<<<gEw, 256, 0, stream>>>(bufA, src, dst, dis, bufB, E);
    k_bias_relu<<<gNH, 256, 0, stream>>>(bufB, b2, totNH);

    // output head
    k_out_softmax<<<gN, 256, 0, stream>>>(bufB, Wout, bout, out, N);
}